// TopKSparseAutoencoder_32289564131763
// MI455X (gfx1250) — compile-verified
//
#include <hip/hip_runtime.h>

// ---------------------------------------------------------------------------
// TopK Sparse Autoencoder forward for MI455X (gfx1250, wave32, WMMA).
//   enc  = (x - bias) @ W_e^T + b_e          dense f32 WMMA GEMM (16x16x4)
//   topk = per-row top-32 mask               radix-select, in-place zeroing
//   dec  = enc_sparse @ W_d^T + b_d + bias   sparse gather over W_e rows
//   mse  = mean((xs - dec)^2)
// Memory-bound (>= 1 GB enc output); f32 WMMA keeps exact precision for free.
// GEMM staging uses gfx1250 async global->LDS copies (ASYNCcnt) with a
// double-buffered tile when the toolchain exposes the builtins.
// ---------------------------------------------------------------------------

static constexpr int kD = 2048;    // embedding size
static constexpr int kF = 16384;   // n_features
static constexpr int kB = 16384;   // batch
static constexpr int kK = 32;      // topk

typedef __attribute__((ext_vector_type(2))) float v2f;
typedef __attribute__((ext_vector_type(8))) float v8f;
typedef int v4i __attribute__((vector_size(16)));   // matches builtin param type

#if defined(__has_builtin)
#  if __has_builtin(__builtin_amdgcn_global_load_async_to_lds_b128) && \
      __has_builtin(__builtin_amdgcn_s_wait_asynccnt)
#    define HAVE_ASYNC_LDS 1
#  endif
#endif
#ifndef HAVE_ASYNC_LDS
#  define HAVE_ASYNC_LDS 0
#endif

#define AS_GLOBAL __attribute__((address_space(1)))
#define AS_LDS    __attribute__((address_space(3)))

#if HAVE_ASYNC_LDS
#define ASYNC_B128(gp, lp) \
  __builtin_amdgcn_global_load_async_to_lds_b128( \
      (AS_GLOBAL v4i*)(void*)(gp), (AS_LDS v4i*)(void*)(lp), 0, 0)
#endif

// ---------------------------------------------------------------------------
// Kernel 1: encoder GEMM, enc[b,f] = sum_d (x[b,d]-bias[d]) * W_e[f,d] + b_e[f]
// Tile: 64(M) x 128(N) per 256-thread block; 8 waves, each wave owns a 16-col
// slab and four 16x16 WMMA C tiles. K moves through a double-buffered LDS tile
// in 16-wide chunks. Tiles are row-major with stride 20 floats (80 B): rows
// stay 16 B aligned for b128 async copies and the stride-20 lattice is LDS
// bank-conflict-free; every fragment is one contiguous 8 B ds_load.
// ---------------------------------------------------------------------------
static constexpr int TM = 64;
static constexpr int TN = 128;
static constexpr int KC = 16;
static constexpr int KP = KC + 4;   // padded row stride (20 floats = 80 B)

__global__ __launch_bounds__(256) void enc_gemm_kernel(
    const float* __restrict__ x, const float* __restrict__ We,
    const float* __restrict__ be, const float* __restrict__ bias,
    float* __restrict__ enc)
{
  __shared__ __align__(16) float As[2][TM][KP];   // x tile (raw; bias applied at frag read)
  __shared__ __align__(16) float Bs[2][TN][KP];   // W_e tile
  __shared__ float Sb[2][KC];                     // bias chunk

  const int t    = threadIdx.x;
  const int lane = t & 31;
  const int wid  = t >> 5;                 // wave id 0..7
  const int row0 = blockIdx.y * TM;
  const int col0 = blockIdx.x * TN;
  const int nw   = wid * 16;               // wave's column slab inside tile

  const int l16   = lane & 15;
  const int khalf = (lane >> 4) << 1;      // ISA A/B layout: lanes16-31 hold K+2

  // staging maps: one b128 per thread for A, two for B (3 async ops per wave)
  const int ar  = t >> 2;                  // A row within tile (0..63)
  const int akq = (t & 3) << 2;            // A k-offset: 0,4,8,12
  const int bc  = t >> 1;                  // B row within tile (0..127)
  const int bkq = (t & 1) << 3;            // B k-offset: 0 or 8 (two b128 each)

  v8f acc[4];
  #pragma unroll
  for (int i = 0; i < 4; ++i)
    #pragma unroll
    for (int v = 0; v < 8; ++v) acc[i][v] = 0.0f;

  auto issue = [&](int k0, int nb) {
#if HAVE_ASYNC_LDS
    // async global->LDS bulk copy, tracked by ASYNCcnt (no VGPR staging)
    ASYNC_B128(&x [(size_t)(row0 + ar) * kD + k0 + akq],     &As[nb][ar][akq]);
    ASYNC_B128(&We[(size_t)(col0 + bc) * kD + k0 + bkq],     &Bs[nb][bc][bkq]);
    ASYNC_B128(&We[(size_t)(col0 + bc) * kD + k0 + bkq + 4], &Bs[nb][bc][bkq + 4]);
#else
    if (k0 + KC < kD)
      __builtin_prefetch(&We[(size_t)(col0 + bc) * kD + k0 + KC + bkq], 0, 1);
    float4 xa = *(const float4*)(&x [(size_t)(row0 + ar) * kD + k0 + akq]);
    float4 w0 = *(const float4*)(&We[(size_t)(col0 + bc) * kD + k0 + bkq]);
    float4 w1 = *(const float4*)(&We[(size_t)(col0 + bc) * kD + k0 + bkq + 4]);
    *(float4*)&As[nb][ar][akq]     = xa;
    *(float4*)&Bs[nb][bc][bkq]     = w0;
    *(float4*)&Bs[nb][bc][bkq + 4] = w1;
#endif
    if (t < KC) Sb[nb][t] = bias[k0 + t];  // tiny chunk via normal path
  };

  auto compute = [&](int nb) {
    #pragma unroll
    for (int ks = 0; ks < 4; ++ks) {        // four K=4 WMMA steps per chunk
      const int kb = (ks << 2) + khalf;
      const v2f sb = *(const v2f*)&Sb[nb][kb];
      const v2f bf = *(const v2f*)&Bs[nb][nw + l16][kb];
      #pragma unroll
      for (int mt = 0; mt < 4; ++mt) {
        v2f af = *(const v2f*)&As[nb][mt * 16 + l16][kb];
        af.x -= sb.x;                        // xs = x - bias, at frag read
        af.y -= sb.y;
        acc[mt] = __builtin_amdgcn_wmma_f32_16x16x4_f32(
            /*neg_a=*/false, af, /*neg_b=*/false, bf,
            /*c_mod=*/(short)0, acc[mt], /*reuse_a=*/false, /*reuse_b=*/false);
      }
    }
  };

  constexpr int NC = kD / KC;   // 128 chunks
  issue(0, 0);
  for (int c = 0; c < NC; ++c) {
    const int cur = c & 1;
    if (c + 1 < NC) issue((c + 1) * KC, cur ^ 1);
#if HAVE_ASYNC_LDS
    // per-wave: drain this wave's chunk-c ops (next chunk's 3 stay in flight);
    // the following barrier then guarantees all waves' chunk-c data landed.
    if (c + 1 < NC) __builtin_amdgcn_s_wait_asynccnt(3);
    else            __builtin_amdgcn_s_wait_asynccnt(0);
#endif
    __syncthreads();            // publish chunk c to all waves
    compute(cur);
    __syncthreads();            // chunk c consumed; its buffer reusable
  }

  // C layout: VGPR v -> lanes 0-15 M=v, lanes 16-31 M=v+8; N = lane&15
  const int mhi = (lane >> 4) << 3;
  #pragma unroll
  for (int mt = 0; mt < 4; ++mt) {
    #pragma unroll
    for (int v = 0; v < 8; ++v) {
      const int m = row0 + mt * 16 + v + mhi;
      const int n = col0 + nw + l16;
      enc[(size_t)m * kF + n] = acc[mt][v] + be[n];
    }
  }
}

// ---------------------------------------------------------------------------
// Kernel 2: per-row top-32 via 4-pass byte radix-select on order-preserving
// key transform; writes values/indices, zeroes non-selected enc entries.
// One 256-thread block per row.
// ---------------------------------------------------------------------------
__device__ __forceinline__ unsigned fkey(float f) {
  unsigned u = __float_as_uint(f);
  return (u & 0x80000000u) ? ~u : (u | 0x80000000u);  // monotone: bigger f -> bigger key
}

__global__ __launch_bounds__(256) void topk_kernel(
    float* __restrict__ enc, float* __restrict__ vals_out, int* __restrict__ idx_out)
{
  const int row = blockIdx.x;
  float* rowp = enc + (size_t)row * kF;
  const int t = threadIdx.x;

  __shared__ unsigned hist[256];
  __shared__ unsigned s_prefix, s_remaining, s_nsel, s_ne;
  __shared__ int      sel_idx[kK];
  __shared__ float    sel_val[kK];

  if (t == 0) { s_prefix = 0u; s_remaining = (unsigned)kK; }
  __syncthreads();

  // ---- radix select: find key T of the 32nd-largest element ----
  for (int p = 3; p >= 0; --p) {
    hist[t] = 0u;
    __syncthreads();
    const unsigned shift  = (unsigned)p * 8u;
    const unsigned himask = (p == 3) ? 0u : (0xFFFFFFFFu << (shift + 8u));
    const unsigned pref   = s_prefix;
    for (int i = t; i < kF; i += 256) {
      const unsigned key = fkey(rowp[i]);
      if ((key & himask) == pref)
        atomicAdd(&hist[(key >> shift) & 255u], 1u);
    }
    __syncthreads();
    if (t == 0) {
      unsigned rem = s_remaining, cum = 0u;
      for (int bb = 255; bb >= 0; --bb) {
        const unsigned c = hist[bb];
        if (cum + c >= rem) {
          s_prefix    = pref | ((unsigned)bb << shift);
          s_remaining = rem - cum;   // ties of this byte still needed
          break;
        }
        cum += c;
      }
    }
    __syncthreads();
  }

  const unsigned T     = s_prefix;      // exact key of the k-th largest
  const unsigned eneed = s_remaining;   // #ties at T to accept
  if (t == 0) { s_nsel = 0u; s_ne = 0u; }
  __syncthreads();

  // ---- gather the 32 selected (value, index) pairs ----
  for (int i = t; i < kF; i += 256) {
    const float v = rowp[i];
    const unsigned key = fkey(v);
    bool take = false;
    if (key > T) take = true;
    else if (key == T) take = (atomicAdd(&s_ne, 1u) < eneed);
    if (take) {
      const unsigned s = atomicAdd(&s_nsel, 1u);
      if (s < (unsigned)kK) { sel_idx[s] = i; sel_val[s] = v; }
    }
  }
  __syncthreads();

  if (t == 0) {  // sort 32 winners descending (top_k output order)
    for (int i = 1; i < kK; ++i) {
      const float v = sel_val[i]; const int id = sel_idx[i];
      int j = i - 1;
      while (j >= 0 && sel_val[j] < v) {
        sel_val[j + 1] = sel_val[j]; sel_idx[j + 1] = sel_idx[j]; --j;
      }
      sel_val[j + 1] = v; sel_idx[j + 1] = id;
    }
  }
  __syncthreads();

  if (t < kK) {
    vals_out[(size_t)row * kK + t] = sel_val[t];
    idx_out [(size_t)row * kK + t] = sel_idx[t];
  }
  __syncthreads();

  // ---- zero all non-selected entries in place ----
  for (int i = t; i < kF; i += 256) {
    const unsigned key = fkey(rowp[i]);
    bool keep = (key > T);
    if (!keep && key == T) {
      #pragma unroll
      for (int j = 0; j < kK; ++j)
        if (sel_idx[j] == i) { keep = true; break; }
    }
    if (!keep) rowp[i] = 0.0f;
  }
}

// ---------------------------------------------------------------------------
// Kernel 3: sparse decode + per-row squared-error partial.
// dec[b,:] = sum_k v[b,k] * W_d[:, idx[b,k]] + b_d + bias.
// W_d == W_e^T by construction, so gather contiguous W_e rows (coalesced;
// 134 MB of W_e sits entirely in the 192 MB L2).
// ---------------------------------------------------------------------------
__global__ __launch_bounds__(256) void decode_kernel(
    const float* __restrict__ x, const float* __restrict__ We,
    const float* __restrict__ bd, const float* __restrict__ bias,
    const float* __restrict__ vals, const int* __restrict__ idxs,
    float* __restrict__ dec, float* __restrict__ partial,
    float* __restrict__ mse_accum, int use_partial)
{
  const int row = blockIdx.x;
  const int t   = threadIdx.x;
  __shared__ float sv[kK];
  __shared__ int   si[kK];
  __shared__ float red[256];

  if (t < kK) {
    sv[t] = vals[(size_t)row * kK + t];
    si[t] = idxs[(size_t)row * kK + t];
  }
  __syncthreads();

  float acc[kD / 256];
  #pragma unroll
  for (int j = 0; j < kD / 256; ++j) acc[j] = 0.0f;

  #pragma unroll 4
  for (int k = 0; k < kK; ++k) {
    const float v = sv[k];
    const float* wr = We + (size_t)si[k] * kD;   // contiguous row of W_e
    #pragma unroll
    for (int j = 0; j < kD / 256; ++j)
      acc[j] += v * wr[t + 256 * j];
  }

  float ss = 0.0f;
  #pragma unroll
  for (int j = 0; j < kD / 256; ++j) {
    const int d = t + 256 * j;
    const float dv = acc[j] + bd[d] + bias[d];
    dec[(size_t)row * kD + d] = dv;
    const float diff = (x[(size_t)row * kD + d] - bias[d]) - dv;
    ss += diff * diff;
  }

  red[t] = ss;
  __syncthreads();
  for (int s = 128; s > 0; s >>= 1) {
    if (t < s) red[t] += red[t + s];
    __syncthreads();
  }
  if (t == 0) {
    if (use_partial) partial[row] = red[0];
    else             atomicAdd(mse_accum, red[0]);
  }
}

// ---------------------------------------------------------------------------
// Kernel 4: finalize mse.
// ---------------------------------------------------------------------------
__global__ __launch_bounds__(256) void mse_finalize_kernel(
    const float* __restrict__ partial, float* __restrict__ mse, int use_partial)
{
  __shared__ float red[256];
  const int t = threadIdx.x;
  const float inv_n = 1.0f / ((float)kB * (float)kD);
  if (use_partial) {
    float s = 0.0f;
    for (int i = t; i < kB; i += 256) s += partial[i];
    red[t] = s;
    __syncthreads();
    for (int k = 128; k > 0; k >>= 1) {
      if (t < k) red[t] += red[t + k];
      __syncthreads();
    }
    if (t == 0) *mse = red[0] * inv_n;
  } else {
    if (t == 0) *mse = *mse * inv_n;   // accumulator already holds the sum
  }
}

__global__ void zero_scalar_kernel(float* p) { if (threadIdx.x == 0) *p = 0.0f; }

// ---------------------------------------------------------------------------
extern "C" void kernel_launch(void* const* d_in, const int* in_sizes, int n_in,
                              void* d_out, int out_size, void* d_ws, size_t ws_size,
                              hipStream_t stream)
{
  const float* x    = (const float*)d_in[0];
  const float* We   = (const float*)d_in[1];
  const float* be   = (const float*)d_in[2];
  const float* Wd   = (const float*)d_in[3];  (void)Wd;  // tied: W_d == W_e^T; decode gathers W_e rows
  const float* bd   = (const float*)d_in[4];
  const float* bias = (const float*)d_in[5];
  (void)in_sizes; (void)n_in; (void)out_size;

  float* out  = (float*)d_out;
  float* enc  = out;                                  // [B, F] masked encoder output
  float* dec  = enc + (size_t)kB * kF;                // [B, D]
  float* mse  = dec + (size_t)kB * kD;                // [1]
  float* vals = mse + 1;                              // [B, K]
  int*   idxs = (int*)(vals + (size_t)kB * kK);       // [B, K]

  const int use_partial = (ws_size >= (size_t)kB * sizeof(float)) ? 1 : 0;
  float* partial = (float*)d_ws;

  // 1) dense encoder GEMM (f32 WMMA), writes pre-mask enc into its final slot
  dim3 g1(kF / TN, kB / TM);
  hipLaunchKernelGGL(enc_gemm_kernel, g1, dim3(256), 0, stream, x, We, be, bias, enc);

  // 2) top-32 per row; masks enc in place, emits values/indices
  hipLaunchKernelGGL(topk_kernel, dim3(kB), dim3(256), 0, stream, enc, vals, idxs);

  // 3) sparse decode + mse partials
  if (!use_partial)
    hipLaunchKernelGGL(zero_scalar_kernel, dim3(1), dim3(32), 0, stream, mse);
  hipLaunchKernelGGL(decode_kernel, dim3(kB), dim3(256), 0, stream,
                     x, We, bd, bias, vals, idxs, dec, partial, mse, use_partial);

  // 4) finalize mse
  hipLaunchKernelGGL(mse_finalize_kernel, dim3(1), dim3(256), 0, stream,
                     partial, mse, use_partial);
}